// CapsClass2d_74448963109577
// MI455X (gfx1250) — compile-verified
//
#include <hip/hip_runtime.h>
#include <math.h>

typedef __attribute__((ext_vector_type(2))) float v2f;
typedef __attribute__((ext_vector_type(8))) float v8f;

#define NB   256      // batch
#define NC   10       // output capsule classes
#define NN   1152     // B*H*W = 32*6*6 input capsules
#define DI   16       // input pose dim
#define DO   16       // output pose dim
#define EPSF 1e-8f

// Async global->LDS copy of 16B (CDNA5, ASYNCcnt-tracked). The instruction
// offset is added to BOTH the LDS and global addresses (CDNA5 ISA
// §4.4), so one address setup covers a whole row with offset: immediates.
#define ASYNC_CP_B128(ldsOff, gptr, imm)                                    \
    asm volatile("global_load_async_to_lds_b128 %0, %1, off offset:" #imm   \
                 :: "v"(ldsOff), "v"(gptr) : "memory")

__device__ __forceinline__ void wait_async_all() {
    asm volatile("s_wait_asynccnt 0x0" ::: "memory");
}

// ---------------------------------------------------------------------------
// Kernel 1: votes[b][c][n][o] = sum_i weight[n][c][o][i] * poses[b][n][i]
// via V_WMMA_F32_16X16X4_F32 (chained 4x for K=16), M=16 batch rows, N=16 o.
// One block per n (1152 blocks), 256 threads = 8 waves; each wave does
// 2 batch-tiles x 10 classes. Tiles staged to LDS with async-to-LDS copies.
// ---------------------------------------------------------------------------
__global__ __launch_bounds__(256)
void caps_votes_wmma_kernel(const float* __restrict__ poses,
                            const float* __restrict__ weight,
                            float* __restrict__ votes) {
    __shared__ float ldsW[NC][DO][DI + 1];   // weight[n][c][o][i]
    __shared__ float ldsP[NB][DI + 1];       // poses[b][n][i]

    const int n   = blockIdx.x;
    const int tid = threadIdx.x;

    // Stage weights for this n: 2560 floats = 640 x 16B async chunks.
    // Chunk j -> c = j/64, o = (j%64)/4, igrp = j%4 (stays inside padded row).
    for (int j = tid; j < (NC * DO * DI) / 4; j += 256) {
        int c = j >> 6;
        int r = j & 63;
        unsigned ldsOff = (unsigned)(uintptr_t)&ldsW[c][r >> 2][(r & 3) * 4];
        const float* g  = weight + (size_t)n * (NC * DO * DI) + (size_t)j * 4;
        ASYNC_CP_B128(ldsOff, g, 0);
    }
    // Stage poses for all 256 b at this n: one 64B row per thread (4 x b128).
    {
        const float*  src    = poses + ((size_t)tid * NN + n) * DI;
        unsigned      ldsOff = (unsigned)(uintptr_t)&ldsP[tid][0];
        ASYNC_CP_B128(ldsOff, src, 0);
        ASYNC_CP_B128(ldsOff, src, 16);
        ASYNC_CP_B128(ldsOff, src, 32);
        ASYNC_CP_B128(ldsOff, src, 48);
    }
    wait_async_all();
    __syncthreads();

    const int  wave  = tid >> 5;
    const int  lane  = tid & 31;
    const int  lrow  = lane & 15;
    const bool hi    = lane >= 16;
    const int  khalf = hi ? 2 : 0;          // lanes 16-31 carry K+2,K+3

    for (int bt = wave; bt < 16; bt += 8) { // 16 batch tiles of 16
        const int b0 = bt * 16;
        // A: 16x4 f32 per step; lane L holds A[M=L][K=khalf..khalf+1]
        v2f afrag[4];
        #pragma unroll
        for (int kk = 0; kk < 4; ++kk) {
            int k = 4 * kk + khalf;
            afrag[kk][0] = ldsP[b0 + lrow][k];
            afrag[kk][1] = ldsP[b0 + lrow][k + 1];
        }
        for (int c = 0; c < NC; ++c) {
            // B: 4x16 f32 per step; B[k][o] = W[o][k]
            v2f bfrag[4];
            #pragma unroll
            for (int kk = 0; kk < 4; ++kk) {
                int k = 4 * kk + khalf;
                bfrag[kk][0] = ldsW[c][lrow][k];
                bfrag[kk][1] = ldsW[c][lrow][k + 1];
            }
            v8f acc = {};
            #pragma unroll
            for (int kk = 0; kk < 4; ++kk) {
                acc = __builtin_amdgcn_wmma_f32_16x16x4_f32(
                        false, afrag[kk], false, bfrag[kk],
                        (short)0, acc, false, false);
            }
            // D layout: VGPR r, lanes0-15 -> D[M=r][N=lane], lanes16-31 -> D[M=8+r][N=lane-16]
            const int mbase = b0 + (hi ? 8 : 0);
            size_t base = (((size_t)mbase * NC + c) * NN + n) * DO + lrow;
            #pragma unroll
            for (int r = 0; r < 8; ++r)
                votes[base + (size_t)r * ((size_t)NC * NN * DO)] = acc[r];
        }
    }
}

// ---------------------------------------------------------------------------
// Kernel 2: dynamic routing, one block per batch element.
// logits + couplings live in LDS (~93 KB), votes streamed from L2/HBM.
// ---------------------------------------------------------------------------
__global__ __launch_bounds__(256)
void caps_routing_kernel(const float* __restrict__ votes,
                         const float* __restrict__ bias,     // [10][16]
                         float* __restrict__ out_poses,      // [256][10][16]
                         float* __restrict__ out_act,        // [256][10]
                         float* __restrict__ out_coup) {     // [256][10][1152]
    __shared__ float logits[NC][NN];
    __shared__ float coup[NC][NN];
    __shared__ float sred[NC][DO];
    __shared__ float vvec[NC][DO];

    const int b   = blockIdx.x;
    const int tid = threadIdx.x;
    const float* vb = votes + (size_t)b * NC * NN * DO;

    for (int j = tid; j < NC * NN; j += 256) (&logits[0][0])[j] = 0.f;
    __syncthreads();

    for (int it = 0; it < 3; ++it) {
        // --- softmax over the class axis at each n ---
        for (int nn = tid; nn < NN; nn += 256) {
            float m = logits[0][nn];
            #pragma unroll
            for (int c = 1; c < NC; ++c) m = fmaxf(m, logits[c][nn]);
            float e[NC], s = 0.f;
            #pragma unroll
            for (int c = 0; c < NC; ++c) { e[c] = __expf(logits[c][nn] - m); s += e[c]; }
            float inv = 1.f / s;
            #pragma unroll
            for (int c = 0; c < NC; ++c) coup[c][nn] = e[c] * inv;
        }
        if (tid < NC * DO) (&sred[0][0])[tid] = bias[tid];
        __syncthreads();

        // --- s[c][o] = bias + sum_n coup[c][n] * votes[c][n][o] ---
        {
            const int grp = tid >> 4;   // 16 groups stride over n
            const int o   = tid & 15;   // lane -> contiguous o, coalesced
            for (int c = 0; c < NC; ++c) {
                float acc = 0.f;
                for (int nn = grp; nn < NN; nn += 16)
                    acc += coup[c][nn] * vb[((size_t)c * NN + nn) * DO + o];
                atomicAdd(&sred[c][o], acc);
            }
        }
        __syncthreads();

        // --- squash ---
        if (tid < NC) {
            const int c = tid;
            float sq = 0.f;
            #pragma unroll
            for (int o = 0; o < DO; ++o) sq += sred[c][o] * sred[c][o];
            float norm  = sqrtf(sq + EPSF);
            float scale = sq / (1.f + sq) / norm;
            #pragma unroll
            for (int o = 0; o < DO; ++o) vvec[c][o] = sred[c][o] * scale;
        }
        __syncthreads();

        // --- logits += v . votes (not on last iteration) ---
        if (it < 2) {
            for (int c = 0; c < NC; ++c) {
                for (int nn = tid; nn < NN; nn += 256) {
                    const float* vp = vb + ((size_t)c * NN + nn) * DO;
                    float dot = 0.f;
                    #pragma unroll
                    for (int o = 0; o < DO; ++o) dot += vvec[c][o] * vp[o];
                    logits[c][nn] += dot;
                }
            }
            __syncthreads();
        }
    }

    // --- outputs ---
    if (tid < NC * DO)
        out_poses[(size_t)b * NC * DO + tid] = (&vvec[0][0])[tid];
    if (tid < NC) {
        float sq = 0.f;
        #pragma unroll
        for (int o = 0; o < DO; ++o) sq += vvec[tid][o] * vvec[tid][o];
        out_act[(size_t)b * NC + tid] = sqrtf(sq + EPSF);
    }
    for (int j = tid; j < NC * NN; j += 256)
        out_coup[(size_t)b * NC * NN + j] = (&coup[0][0])[j];
}

// ---------------------------------------------------------------------------
extern "C" void kernel_launch(void* const* d_in, const int* in_sizes, int n_in,
                              void* d_out, int out_size, void* d_ws, size_t ws_size,
                              hipStream_t stream) {
    (void)in_sizes; (void)n_in; (void)out_size; (void)ws_size;
    const float* poses  = (const float*)d_in[0];   // (256,32,6,6,16,1) f32
    // d_in[1] = input_caps_activations (unused by the reference math)
    const float* weight = (const float*)d_in[2];   // (32,6,6,10,16,16) f32
    const float* bias   = (const float*)d_in[3];   // (10,1,1,16,1)     f32

    float* votes = (float*)d_ws;                   // [256][10][1152][16] f32 (~180 MiB)

    float* out_poses = (float*)d_out;                          // 40960 floats
    float* out_act   = out_poses + (size_t)NB * NC * DO;       // 2560 floats
    float* out_coup  = out_act   + (size_t)NB * NC;            // 2949120 floats

    caps_votes_wmma_kernel<<<NN, 256, 0, stream>>>(poses, weight, votes);
    caps_routing_kernel<<<NB, 256, 0, stream>>>(votes, bias, out_poses, out_act, out_coup);
}